// MdsOWLoss_73770358276630
// MI455X (gfx1250) — compile-verified
//
#include <hip/hip_runtime.h>
#include <hip/hip_bf16.h>

// Problem dims (fixed by setup_inputs): B=4, K=256, D=256, H=W=256
#define EPS_   1e-8f
#define BATCH_ 4
#define K_     256
#define D_     256
#define HW_    65536            // 256*256
#define NPIX_  262144           // BATCH_*HW_

typedef float v2f __attribute__((ext_vector_type(2)));
typedef float v8f __attribute__((ext_vector_type(8)));

// ---------------------------------------------------------------------------
// ws layout (bytes):
//   [0        , 1048576)  sem  : int[NPIX_]
//   [1048576  , 1572864)  tab  : float2[K_*D_]   (mean, keep ? 1/(var+eps) : 0)
//   [1572864  , 1576960)  cntA : int[1024]       per-block keep counts
//   [1576960  , 1585152)  sumB : float[2048]     per-block partial L1 sums
// ---------------------------------------------------------------------------

__global__ void mds_build_table(const float* __restrict__ pf,
                                const float* __restrict__ var,
                                const int*   __restrict__ pc,
                                float2*      __restrict__ tab) {
    int i = blockIdx.x * blockDim.x + threadIdx.x;   // 0 .. K_*D_-1
    int k = i >> 8;                                  // D_ == 256
    float inv = (pc[k] > 0) ? (1.0f / (var[i] + EPS_)) : 0.0f;
    tab[i] = make_float2(pf[i], inv);
}

__global__ void mds_argmax(const float* __restrict__ logits,
                           const int*   __restrict__ pc,
                           int*         __restrict__ sem,
                           int*         __restrict__ cntA) {
    __shared__ int lds[256];
    int n  = blockIdx.x * 256 + threadIdx.x;         // pixel id
    int b  = n >> 16;                                // HW_ == 65536
    int hw = n & (HW_ - 1);
    const float* lb = logits + (size_t)b * K_ * HW_ + hw;

    float best = lb[0];
    int   bi   = 0;
#pragma unroll 4
    for (int k = 1; k < K_; ++k) {
        // help the 256KB-strided stream along (global_prefetch_b8)
        if (k + 8 < K_) __builtin_prefetch(lb + (size_t)(k + 8) * HW_, 0, 1);
        float v = lb[(size_t)k * HW_];
        if (v > best) { best = v; bi = k; }          // strict > == first-max (argmax)
    }
    sem[n] = bi;

    lds[threadIdx.x] = (pc[bi] > 0) ? 1 : 0;
    __syncthreads();
    for (int s = 128; s > 0; s >>= 1) {
        if (threadIdx.x < s) lds[threadIdx.x] += lds[threadIdx.x + s];
        __syncthreads();
    }
    if (threadIdx.x == 0) cntA[blockIdx.x] = lds[0];
}

// Wave32: 16 pixels per wave; D-reduction via V_WMMA_F32_16X16X4_F32 with
// B = ones, so C row m accumulates sum_d |f(m,d)-mean| * inv(m,d).
__global__ void mds_l1sum(const float*  __restrict__ ue,
                          const int*    __restrict__ sem,
                          const float2* __restrict__ tab,
                          float*        __restrict__ sumB) {
    __shared__ float wsum[8];
    const int lane = threadIdx.x & 31;
    const int wv   = threadIdx.x >> 5;               // 8 waves / block
    const int n    = blockIdx.x * 128 + wv * 16 + (lane & 15);
    const int ko   = (lane >> 4) << 1;               // 0 for lanes 0-15, 2 for 16-31
    const int b    = n >> 16;
    const int hw   = n & (HW_ - 1);

    const float*  fb = ue + (size_t)b * D_ * HW_ + hw;   // + d*HW_ per feature
    const int     s  = sem[n];
    const float2* tr = tab + (size_t)s * D_;

    v8f c    = {0.f, 0.f, 0.f, 0.f, 0.f, 0.f, 0.f, 0.f};
    v2f ones = {1.f, 1.f};

    for (int d0 = 0; d0 < D_; d0 += 4) {
        const int d = d0 + ko;
        float  f0 = fb[(size_t)d * HW_];
        float  f1 = fb[(size_t)(d + 1) * HW_];
        float2 t0 = tr[d];
        float2 t1 = tr[d + 1];
        v2f a;
        a.x = fabsf(f0 - t0.x) * t0.y;               // A[m][ko]
        a.y = fabsf(f1 - t1.x) * t1.y;               // A[m][ko+1]
        // D = A(16x4) * ones(4x16) + C  -> every column holds per-pixel row sums
        c = __builtin_amdgcn_wmma_f32_16x16x4_f32(
                false, a, false, ones, (short)0, c, false, false);
    }

    // lane j (<16): c[v] = rowsum(pixel v); lane j+16: c[v] = rowsum(pixel v+8)
    float srow = c[0] + c[1] + c[2] + c[3] + c[4] + c[5] + c[6] + c[7];
    float tot  = __shfl(srow, 0, 32) + __shfl(srow, 16, 32);

    if (lane == 0) wsum[wv] = tot;
    __syncthreads();
    if (threadIdx.x == 0) {
        float acc = 0.f;
        for (int i = 0; i < 8; ++i) acc += wsum[i];  // fixed order -> deterministic
        sumB[blockIdx.x] = acc;
    }
}

__global__ void mds_finalize(const float* __restrict__ sumB,
                             const int*   __restrict__ cntA,
                             float*       __restrict__ out) {
    __shared__ float fs[256];
    __shared__ int   is[256];
    float a = 0.f;
    for (int i = threadIdx.x; i < 2048; i += 256) a += sumB[i];
    int ci = 0;
    for (int i = threadIdx.x; i < 1024; i += 256) ci += cntA[i];
    fs[threadIdx.x] = a;
    is[threadIdx.x] = ci;
    __syncthreads();
    for (int s = 128; s > 0; s >>= 1) {
        if (threadIdx.x < s) {
            fs[threadIdx.x] += fs[threadIdx.x + s];
            is[threadIdx.x] += is[threadIdx.x + s];
        }
        __syncthreads();
    }
    if (threadIdx.x == 0) out[0] = fs[0] / ((float)is[0] * (float)D_);
}

extern "C" void kernel_launch(void* const* d_in, const int* in_sizes, int n_in,
                              void* d_out, int out_size, void* d_ws, size_t ws_size,
                              hipStream_t stream) {
    const float* ue     = (const float*)d_in[0];   // (B, D, H, W)
    const float* logits = (const float*)d_in[1];   // (B, K, H, W)
    const float* pf     = (const float*)d_in[2];   // (K, D)
    const float* var    = (const float*)d_in[3];   // (K, D)
    const int*   pc     = (const int*)d_in[4];     // (K,)
    float*       out    = (float*)d_out;

    char*   ws   = (char*)d_ws;
    int*    sem  = (int*)ws;                       // 1 MB
    float2* tab  = (float2*)(ws + 1048576);        // 512 KB
    int*    cntA = (int*)(ws + 1572864);           // 4 KB
    float*  sumB = (float*)(ws + 1576960);         // 8 KB

    mds_build_table<<<K_ * D_ / 256, 256, 0, stream>>>(pf, var, pc, tab);
    mds_argmax     <<<NPIX_ / 256,   256, 0, stream>>>(logits, pc, sem, cntA);
    mds_l1sum      <<<NPIX_ / 128,   256, 0, stream>>>(ue, sem, tab, sumB);
    mds_finalize   <<<1,             256, 0, stream>>>(sumB, cntA, out);
}